// EdgeDecoder_81071802679525
// MI455X (gfx1250) — compile-verified
//
#include <hip/hip_runtime.h>

typedef __attribute__((ext_vector_type(2))) float v2f;
typedef __attribute__((ext_vector_type(8))) float v8f;

#define HID 8     // hidden size
#define HID2 16   // concat size

// D = A(16x4) x B(4x16) + C, chained over 4 K-steps to get K=16.
//   A = W1^T  : A[M=out][K=in], rows M>=8 are zero
//   B = Z^T   : B[K=feature][N=edge]  (features 0-7 from z_demand[row], 8-15 from z_measurement[col])
//   D[M=out][N=edge] layout: VGPR i, lane l -> M = i + 8*(l>=16), N = l%16
// => lanes 0-15: accumulator VGPR i holds hidden channel i of edge (lane). Layer-2 is per-lane.
//
// Addressing: all offsets fit in 32 bits (E=16M edges -> <=256MB into eli;
// node tables 3.2MB; output 64MB), so use unsigned element offsets to get
// saddr64 + vgpr32 global addressing instead of 64-bit VGPR-pair arithmetic.
__global__ __launch_bounds__(256) void edge_decoder_wmma_f32(
    const float* __restrict__ zd,        // [N_NODES, 8]
    const float* __restrict__ zm,        // [N_NODES, 8]
    const long long* __restrict__ eli,   // [2, E] int64
    const float* __restrict__ w1,        // [16, 8] row-major (in, out)
    const float* __restrict__ b1,        // [8]
    const float* __restrict__ w2,        // [8]
    const float* __restrict__ b2,        // [1]
    float* __restrict__ out,             // [E]
    const long long E)
{
    const int lane = threadIdx.x & 31;
    const int m8   = lane & 15;          // M for A / N (edge-in-tile) for B
    const int lg   = lane >> 4;          // half-wave: K offset +2 for hi half
    const unsigned Eu = (unsigned)E;

    // ---- preload A = W1^T chunks (one per K-step), biases, w2 (amortized over loop) ----
    v2f A[4];
#pragma unroll
    for (int kk = 0; kk < 4; ++kk) {
        const int K0 = 4 * kk + 2 * lg;
        v2f a;
        a[0] = (m8 < HID) ? w1[(K0 + 0) * HID + m8] : 0.0f;  // A[M=m8][K=K0]
        a[1] = (m8 < HID) ? w1[(K0 + 1) * HID + m8] : 0.0f;  // A[M=m8][K=K0+1]
        A[kk] = a;
    }
    float bias1[HID], wt2[HID];
#pragma unroll
    for (int i = 0; i < HID; ++i) { bias1[i] = b1[i]; wt2[i] = w2[i]; }
    const float bias2 = b2[0];

    const long long nPairs = (E + 31) >> 5;  // 32 edges (2 WMMA tiles) per iteration
    const long long waveId = (((long long)blockIdx.x * blockDim.x) + threadIdx.x) >> 5;
    const long long nWaves = ((long long)gridDim.x * blockDim.x) >> 5;

    for (long long p = waveId; p < nPairs; p += nWaves) {
        const unsigned base = (unsigned)p << 5;          // edge index of this 32-edge pair
        const bool fullPair = (base + 32u) <= Eu;        // wave-uniform tail check
        v8f accs[2];
        float sig[2];

#pragma unroll
        for (int t = 0; t < 2; ++t) {
            unsigned e = base + (unsigned)(t * 16 + m8);
            if (!fullPair && e >= Eu) e = Eu - 1u;       // tail only; store is guarded
            const long long ri = eli[e];                 // demand index (int64 value)
            const long long ci = eli[Eu + e];            // measurement index
            const unsigned roff = (unsigned)ri << 5;     // *8 floats *4 bytes (32-bit path)
            const unsigned coff = (unsigned)ci << 5;
            const char* prow = (const char*)zd + roff;
            const char* pcol = (const char*)zm + coff;

            v8f acc = {};
#pragma unroll
            for (int kk = 0; kk < 4; ++kk) {
                const int f = 4 * kk + 2 * lg;           // feature pair for this lane/K-step
                const char* src = (f < HID) ? (prow + 4 * f) : (pcol + 4 * (f - HID));
                const v2f b = *(const v2f*)src;          // 8B gather, L2-resident tables
                acc = __builtin_amdgcn_wmma_f32_16x16x4_f32(
                    false, A[kk], false, b, (short)0, acc, false, false);
            }
            accs[t] = acc;

            // ---- layer 2 per lane: logit = b2 + sum_i relu(h_i + b1_i) * w2_i ----
            float a2 = bias2;
#pragma unroll
            for (int i = 0; i < HID; ++i) {
                float h = accs[t][i] + bias1[i];
                h = h > 0.0f ? h : 0.0f;
                a2 += h * wt2[i];
            }
            // sigmoid via v_exp_f32 + v_rcp_f32 (no IEEE div sequence)
            sig[t] = __builtin_amdgcn_rcpf(1.0f + __expf(-a2));
        }

        // lanes 0-15 hold tile0 results; pull tile1 results into lanes 16-31 for a
        // single coalesced 32-lane store.
        const float r1 = __shfl(sig[1], m8, 32);
        const float res = (lg == 0) ? sig[0] : r1;
        const unsigned eo = base + (unsigned)lane;
        if (eo < Eu) out[eo] = res;
    }
}

extern "C" void kernel_launch(void* const* d_in, const int* in_sizes, int n_in,
                              void* d_out, int out_size, void* d_ws, size_t ws_size,
                              hipStream_t stream) {
    const float*     zd  = (const float*)d_in[0];
    const float*     zm  = (const float*)d_in[1];
    const long long* eli = (const long long*)d_in[2];   // int64 [2, E]
    const float*     w1  = (const float*)d_in[3];
    const float*     b1  = (const float*)d_in[4];
    const float*     w2  = (const float*)d_in[5];
    const float*     b2  = (const float*)d_in[6];
    float*           out = (float*)d_out;

    const long long E = (long long)in_sizes[2] / 2;

    const int threads = 256;                 // 8 wave32 per block
    const long long nPairs = (E + 31) >> 5;
    long long blocksNeeded = (nPairs * 32 + threads - 1) / threads;
    int blocks = (int)(blocksNeeded < 4096 ? blocksNeeded : 4096);
    if (blocks < 1) blocks = 1;

    edge_decoder_wmma_f32<<<blocks, threads, 0, stream>>>(
        zd, zm, eli, w1, b1, w2, b2, out, E);
}